// Bilinear_54065048322517
// MI455X (gfx1250) — compile-verified
//
#include <hip/hip_runtime.h>
#include <hip/hip_bf16.h>

typedef float v2f __attribute__((ext_vector_type(2)));
typedef float v8f __attribute__((ext_vector_type(8)));

#define B_SZ 16384
#define I_SZ 64
#define J_SZ 2048
#define K_SZ 64

// LDS layout: row p (p = k/2, 0..31) holds pairs (w2[2p][j], w2[2p+1][j]) for the
// block's 64-column J slice. Row stride 80 float2 (=160 dwords ≡ 32 mod 64 banks)
// so lane-groups 0-15 and 16-31 (which read rows 2t and 2t+1) hit disjoint bank halves.
#define LW_STRIDE 80

// ---------------------------------------------------------------------------
// Kernel 1: w2[i][j] = sum_k weights[i][j][k]   (K contiguous, float4 loads)
// ---------------------------------------------------------------------------
__global__ __launch_bounds__(256) void reduce_w_kernel(const float* __restrict__ w,
                                                       float* __restrict__ w2) {
    const int idx = blockIdx.x * 256 + threadIdx.x;          // 0 .. 64*2048-1
    const float4* p = (const float4*)(w + (size_t)idx * K_SZ);
    float s = 0.0f;
#pragma unroll
    for (int t = 0; t < K_SZ / 4; ++t) {
        float4 v = p[t];
        s += (v.x + v.y) + (v.z + v.w);
    }
    w2[idx] = s;
}

// ---------------------------------------------------------------------------
// Kernel 2: out[b][j] = (sum_i x1[b][i] * w2[i][j]) * x2[b][j]
// Block = 8 waves; block owns a 64-wide J slice (w2 slice staged in LDS);
// each wave computes a 16x64 output strip with v_wmma_f32_16x16x4_f32.
// ---------------------------------------------------------------------------
__global__ __launch_bounds__(256) void bilinear_gemm_kernel(
    const float* __restrict__ x1,   // (B, 64)
    const float* __restrict__ x2,   // (B, 2048)
    const float* __restrict__ w2,   // (64, 2048)
    float* __restrict__ out)        // (B, 2048)
{
    __shared__ v2f lw[32 * LW_STRIDE];

    const int lane = threadIdx.x & 31;
    const int wave = threadIdx.x >> 5;

    const int colg   = blockIdx.x & 31;        // J/64 slice index
    const int rowblk = blockIdx.x >> 5;        // 0..127
    const int j0 = colg * 64;
    const int b0 = (rowblk * 8 + wave) * 16;   // 16 rows per wave

    // ---- cooperative LDS fill of w2 slice: pairs (k=2p, k=2p+1) interleaved ----
#pragma unroll
    for (int r = 0; r < 8; ++r) {
        const int idx = threadIdx.x + 256 * r; // 0..2047
        const int p   = idx >> 6;              // 0..31
        const int jj  = idx & 63;
        v2f v;
        v.x = w2[(size_t)(2 * p + 0) * J_SZ + j0 + jj];
        v.y = w2[(size_t)(2 * p + 1) * J_SZ + j0 + jj];
        lw[p * LW_STRIDE + jj] = v;
    }
    __syncthreads();

    const int m = lane & 15;   // A row / B-C column within tile
    const int g = lane >> 4;   // lane group: selects K-high (A/B) or M-high (C/D)

    // ---- A fragments: ISA 16x4 fp32 A layout ----
    // k-step t covers K = 4t..4t+3; lane group g supplies K = 4t+2g, 4t+2g+1.
    // Contiguous pair in x1 row -> 8-byte aligned float2 global loads.
    v2f afrag[16];
    const float* arow = x1 + (size_t)(b0 + m) * I_SZ + 2 * g;
#pragma unroll
    for (int t = 0; t < 16; ++t) {
        afrag[t] = *(const v2f*)(arow + 4 * t);
    }

    v8f acc[4];
#pragma unroll
    for (int c = 0; c < 4; ++c) acc[c] = (v8f){0.f, 0.f, 0.f, 0.f, 0.f, 0.f, 0.f, 0.f};

    // ---- 16 K-steps x 4 column tiles: 64 WMMAs, A reused across tiles ----
#pragma unroll
    for (int t = 0; t < 16; ++t) {
        const v2f* brow = &lw[(2 * t + g) * LW_STRIDE + m];
#pragma unroll
        for (int c = 0; c < 4; ++c) {
            v2f bfrag = brow[16 * c];   // {B[4t+2g][n], B[4t+2g+1][n]}
            acc[c] = __builtin_amdgcn_wmma_f32_16x16x4_f32(
                /*neg_a=*/false, afrag[t],
                /*neg_b=*/false, bfrag,
                /*c_mod=*/(short)0, acc[c],
                /*reuse_a=*/false, /*reuse_b=*/false);
        }
    }

    // ---- epilogue: C/D layout (VGPR v -> rows v, v+8; lane -> column) ----
#pragma unroll
    for (int c = 0; c < 4; ++c) {
        const size_t base = (size_t)(b0 + 8 * g) * J_SZ + j0 + 16 * c + m;
#pragma unroll
        for (int v = 0; v < 8; ++v) {
            const size_t off = base + (size_t)v * J_SZ;
            out[off] = acc[c][v] * x2[off];
        }
    }
}

// ---------------------------------------------------------------------------
extern "C" void kernel_launch(void* const* d_in, const int* in_sizes, int n_in,
                              void* d_out, int out_size, void* d_ws, size_t ws_size,
                              hipStream_t stream) {
    const float* x1 = (const float*)d_in[0];   // (16384, 64)
    const float* x2 = (const float*)d_in[1];   // (16384, 2048)
    const float* w  = (const float*)d_in[2];   // (64, 2048, 64)
    float* out = (float*)d_out;                // (16384, 2048)
    float* w2  = (float*)d_ws;                 // scratch: 64*2048 floats = 512 KiB

    // Step 1: reduce weights over K
    reduce_w_kernel<<<(I_SZ * J_SZ) / 256, 256, 0, stream>>>(w, w2);

    // Step 2: fused GEMM (K=64) + elementwise multiply
    // grid = (B/16/8 row-blocks) * (J/64 col-slices) = 128 * 32 = 4096
    bilinear_gemm_kernel<<<(B_SZ / 16 / 8) * (J_SZ / 64), 256, 0, stream>>>(x1, x2, w2, out);
}